// GraphConvolution_52587579572945
// MI455X (gfx1250) — compile-verified
//
#include <hip/hip_runtime.h>

// GCN layer for MI455X (gfx1250, wave32):
//   h = x @ w            (WMMA f32 16x16x4; x tile staged via
//                         GLOBAL_LOAD_ASYNC_TO_LDS_B128 + s_wait_asynccnt)
//   out = b              (bias broadcast)
//   out[r] += vals[e]*h[c]   (COO scatter via global_atomic_add_f32, L2-resident h)
//   out = relu(out)

#define F_IN 256
#define F_OUT 128
#define LDS_STRIDE 260   // 256 + 4 floats pad: row stride not 0 mod 64 banks
#define MROWS 32         // rows of x per block (2 M-tiles per wave)

typedef float v2f __attribute__((ext_vector_type(2)));
typedef float v8f __attribute__((ext_vector_type(8)));

// ---------------------------------------------------------------------------
// Kernel 1: h = x @ w via V_WMMA_F32_16X16X4_F32.
// Block = 256 threads = 8 waves. Block owns 32 rows of x; wave w owns two
// 16x16 output tiles (rows [0,16) and [16,32) of the block strip) at
// columns [16w, 16w+16). B fragment is loaded once per k-step, used twice.
// ---------------------------------------------------------------------------
__global__ __launch_bounds__(256) void gcn_gemm_wmma(
    const float* __restrict__ x, const float* __restrict__ w,
    float* __restrict__ h, int n_rows)
{
  __shared__ float xs[MROWS * LDS_STRIDE];

  const int t    = threadIdx.x;
  const int lane = t & 31;
  const int wave = t >> 5;
  const int rowBase = blockIdx.x * MROWS;

  // --- Async fill: 32x256 f32 strip of x -> LDS, B128 per lane, no VGPR
  // round-trip. ASYNCcnt tracks completion; padded rows keep 16B alignment
  // (row stride 1040 B).
  for (int j = 0; j < 8; ++j) {
    int lin = t + j * 256;            // float4 index; 64 float4 per row
    int r   = lin >> 6;
    int c4  = lin & 63;
    int gr  = rowBase + r;
    if (gr >= n_rows) gr = n_rows - 1;  // tail clamp (N=100000 is exact)
    const float* gp = x + (size_t)gr * F_IN + c4 * 4;
    unsigned lp = (unsigned)(unsigned long long)(const void*)&xs[r * LDS_STRIDE + c4 * 4];
    asm volatile("global_load_async_to_lds_b128 %0, %1, off"
                 :: "v"(lp), "v"(gp) : "memory");
  }
  asm volatile("s_wait_asynccnt 0x0" ::: "memory");
  __syncthreads();

  // A 16x4 f32 layout: lanes 0-15 -> M=lane, K={k,k+1}; lanes 16-31 -> K={k+2,k+3}
  // B 4x16 f32 layout: lanes 0-15 -> N=lane, K={k,k+1}; lanes 16-31 -> K={k+2,k+3}
  const int m     = lane & 15;
  const int khalf = (lane >> 4) << 1;        // 0 or 2
  const int n     = (wave << 4) + (lane & 15);

  v8f acc0 = {};
  v8f acc1 = {};
  #pragma unroll 4
  for (int k = 0; k < F_IN; k += 4) {
    v2f b;
    b.x = w[(size_t)(k + khalf)     * F_OUT + n];   // coalesced 64B per half-wave
    b.y = w[(size_t)(k + khalf + 1) * F_OUT + n];
    const float2 a0v =
        *reinterpret_cast<const float2*>(&xs[m * LDS_STRIDE + k + khalf]);
    const float2 a1v =
        *reinterpret_cast<const float2*>(&xs[(m + 16) * LDS_STRIDE + k + khalf]);
    v2f a0; a0.x = a0v.x; a0.y = a0v.y;
    v2f a1; a1.x = a1v.x; a1.y = a1v.y;
    // (neg_a, A, neg_b, B, c_mod, C, reuse_a, reuse_b)
    acc0 = __builtin_amdgcn_wmma_f32_16x16x4_f32(false, a0, false, b,
                                                 (short)0, acc0, false, false);
    acc1 = __builtin_amdgcn_wmma_f32_16x16x4_f32(false, a1, false, b,
                                                 (short)0, acc1, false, false);
  }

  // C/D layout: VGPR v holds row v (lanes 0-15) / row v+8 (lanes 16-31), col=lane%16.
  const int rOff = (lane >> 4) << 3;
  const int col  = (wave << 4) + (lane & 15);
  float* hp0 = h + ((size_t)rowBase + rOff) * F_OUT + col;
  float* hp1 = hp0 + (size_t)16 * F_OUT;

  if (rowBase + MROWS <= n_rows) {
    // fast path: no guards, straight-line stores
    #pragma unroll
    for (int v = 0; v < 8; ++v) hp0[(size_t)v * F_OUT] = acc0[v];
    #pragma unroll
    for (int v = 0; v < 8; ++v) hp1[(size_t)v * F_OUT] = acc1[v];
  } else {
    #pragma unroll
    for (int v = 0; v < 8; ++v)
      if (rowBase + rOff + v < n_rows) hp0[(size_t)v * F_OUT] = acc0[v];
    #pragma unroll
    for (int v = 0; v < 8; ++v)
      if (rowBase + 16 + rOff + v < n_rows) hp1[(size_t)v * F_OUT] = acc1[v];
  }
}

// ---------------------------------------------------------------------------
// Kernel 2: out[n][f] = b[f]  (d_out is poisoned; must init before atomics)
// ---------------------------------------------------------------------------
__global__ __launch_bounds__(256) void gcn_bias_init(
    float* __restrict__ out, const float* __restrict__ b, int total)
{
  int i = blockIdx.x * 256 + threadIdx.x;
  if (i < total) out[i] = b[i & (F_OUT - 1)];
}

// ---------------------------------------------------------------------------
// Kernel 3: COO scatter. 32 lanes per edge, 4 features per lane (float4
// gather of h, which is fully L2-resident at 51 MB vs 192 MB L2).
// ---------------------------------------------------------------------------
__global__ __launch_bounds__(256) void gcn_edge_scatter(
    const int* __restrict__ adj_row, const int* __restrict__ adj_col,
    const float* __restrict__ adj_vals, const float* __restrict__ h,
    float* __restrict__ out, long long total)  // total = E*32
{
  long long idx = (long long)blockIdx.x * 256 + threadIdx.x;
  if (idx >= total) return;
  const int e  = (int)(idx >> 5);
  const int fq = ((int)idx & 31) << 2;   // feature quad base [0,128)

  const int   c = adj_col[e];
  const int   r = adj_row[e];
  const float v = adj_vals[e];

  const float4 hv =
      *reinterpret_cast<const float4*>(h + (size_t)c * F_OUT + fq);
  float* o = out + (size_t)r * F_OUT + fq;
  atomicAdd(o + 0, v * hv.x);
  atomicAdd(o + 1, v * hv.y);
  atomicAdd(o + 2, v * hv.z);
  atomicAdd(o + 3, v * hv.w);
}

// ---------------------------------------------------------------------------
// Kernel 4: out = relu(out)
// ---------------------------------------------------------------------------
__global__ __launch_bounds__(256) void gcn_relu(float* __restrict__ out,
                                                int total)
{
  int i = blockIdx.x * 256 + threadIdx.x;
  if (i < total) out[i] = fmaxf(out[i], 0.0f);
}

// ---------------------------------------------------------------------------
extern "C" void kernel_launch(void* const* d_in, const int* in_sizes, int n_in,
                              void* d_out, int out_size, void* d_ws,
                              size_t ws_size, hipStream_t stream)
{
  const float* x        = (const float*)d_in[0];
  const int*   adj_row  = (const int*)  d_in[1];
  const int*   adj_col  = (const int*)  d_in[2];
  const float* adj_vals = (const float*)d_in[3];
  const float* w        = (const float*)d_in[4];
  const float* b        = (const float*)d_in[5];

  const int N = in_sizes[0] / F_IN;
  const int E = in_sizes[1];

  float* out = (float*)d_out;
  float* h   = (float*)d_ws;          // N*F_OUT floats = 51.2 MB scratch

  // 1) h = x @ w
  const int rowTiles = (N + MROWS - 1) / MROWS;
  gcn_gemm_wmma<<<rowTiles, 256, 0, stream>>>(x, w, h, N);

  // 2) out = b
  const int total = N * F_OUT;
  gcn_bias_init<<<(total + 255) / 256, 256, 0, stream>>>(out, b, total);

  // 3) scatter edges
  const long long etotal = (long long)E * 32;
  const int eblocks = (int)((etotal + 255) / 256);
  gcn_edge_scatter<<<eblocks, 256, 0, stream>>>(adj_row, adj_col, adj_vals,
                                                h, out, etotal);

  // 4) relu
  gcn_relu<<<(total + 255) / 256, 256, 0, stream>>>(out, total);
}